// MANNCell_90555090468988
// MI455X (gfx1250) — compile-verified
//
#include <hip/hip_runtime.h>
#include <hip/hip_bf16.h>

#define BB   256   // batch
#define HH   1024  // hidden
#define MM   256   // mem slots
#define HSD  128   // head size (HS)
#define N3H  3072  // 3*H

typedef float v8f __attribute__((ext_vector_type(8)));
typedef float v2f __attribute__((ext_vector_type(2)));
typedef float f4  __attribute__((ext_vector_type(4)));

__device__ __forceinline__ v8f wmma4(v2f a, v2f b, v8f c) {
  // V_WMMA_F32_16X16X4_F32 : D = A(16x4) * B(4x16) + C(16x16)
  return __builtin_amdgcn_wmma_f32_16x16x4_f32(false, a, false, b, (short)0, c,
                                               false, false);
}

__device__ __forceinline__ float sigmoidf_(float x) { return 1.0f / (1.0f + expf(-x)); }
__device__ __forceinline__ float gumbelf_(float u) {
  return -logf(1e-20f - logf(1e-20f + u));
}

// ---------------------------------------------------------------------------
// Generic fused GEMM: Out = act( A1@W1 [+ A2@W2 [+ A3@W3]] + bias )
// A*: (256 x K) row-major, stride lda.  W*: (K x N) row-major, stride N.
// One wave (32 threads) computes a 16(M) x 64(N) tile via WMMA f32 16x16x4.
// grid = (ceil(N/64), 256/16), blockDim = 32.
// ACT: 0=none, 1=sigmoid, 2=tanh
// ---------------------------------------------------------------------------
template<int NA, int ACT>
__global__ void gemm3_wmma(const float* __restrict__ A1, const float* __restrict__ W1,
                           const float* __restrict__ A2, const float* __restrict__ W2,
                           const float* __restrict__ A3, const float* __restrict__ W3,
                           const float* __restrict__ bias, float* __restrict__ Out,
                           int K, int N, int lda)
{
  const int lane = threadIdx.x & 31;
  const int half = lane >> 4;          // 0: K pair {0,1} / rows 0..7 ; 1: K pair {2,3} / rows 8..15
  const int r    = lane & 15;
  const int m0   = blockIdx.y * 16;
  const int n0   = blockIdx.x * 64;

  v8f acc[4] = {};
  int  ncol[4];
  bool nval[4];
#pragma unroll
  for (int t = 0; t < 4; ++t) {
    int n = n0 + t * 16 + r;
    nval[t] = (n < N);
    ncol[t] = nval[t] ? n : 0;         // clamp: loads stay in-bounds, result masked at store
  }

  const int arow = m0 + r;             // B=256, grid.y*16 covers exactly
  const float* a1p = A1 + (size_t)arow * lda + 2 * half;
  const float* a2p = (NA >= 2) ? A2 + (size_t)arow * lda + 2 * half : nullptr;
  const float* a3p = (NA >= 3) ? A3 + (size_t)arow * lda + 2 * half : nullptr;

  for (int k0 = 0; k0 < K; k0 += 4) {
    v2f a1, a2, a3;
    a1.x = a1p[k0]; a1.y = a1p[k0 + 1];
    if (NA >= 2) { a2.x = a2p[k0]; a2.y = a2p[k0 + 1]; }
    if (NA >= 3) { a3.x = a3p[k0]; a3.y = a3p[k0 + 1]; }

    const int krow = k0 + 2 * half;
    {
      const float* w0 = W1 + (size_t)krow * N;
      const float* w1 = w0 + N;
#pragma unroll
      for (int t = 0; t < 4; ++t) {
        v2f b; b.x = w0[ncol[t]]; b.y = w1[ncol[t]];
        acc[t] = wmma4(a1, b, acc[t]);
      }
    }
    if (NA >= 2) {
      const float* w0 = W2 + (size_t)krow * N;
      const float* w1 = w0 + N;
#pragma unroll
      for (int t = 0; t < 4; ++t) {
        v2f b; b.x = w0[ncol[t]]; b.y = w1[ncol[t]];
        acc[t] = wmma4(a2, b, acc[t]);
      }
    }
    if (NA >= 3) {
      const float* w0 = W3 + (size_t)krow * N;
      const float* w1 = w0 + N;
#pragma unroll
      for (int t = 0; t < 4; ++t) {
        v2f b; b.x = w0[ncol[t]]; b.y = w1[ncol[t]];
        acc[t] = wmma4(a3, b, acc[t]);
      }
    }
  }

#pragma unroll
  for (int t = 0; t < 4; ++t) {
    if (!nval[t]) continue;
    const int col = n0 + t * 16 + r;
    const float bb = bias ? bias[col] : 0.0f;
#pragma unroll
    for (int v = 0; v < 8; ++v) {
      const int row = m0 + v + 8 * half;
      float val = acc[t][v] + bb;
      if (ACT == 1) val = sigmoidf_(val);
      else if (ACT == 2) val = tanhf(val);
      Out[(size_t)row * N + col] = val;
    }
  }
}

// ---------------------------------------------------------------------------
// pbias[b] = sum_s head[b, s] * fc1_b[s]   (s < 128, head stride 129)
// blockDim=256 (8 waves), one wave per b; grid = 32.
// ---------------------------------------------------------------------------
__global__ void proj_bias(const float* __restrict__ head, const float* __restrict__ fc1_b,
                          float* __restrict__ pbias)
{
  const int lane = threadIdx.x & 31;
  const int b = blockIdx.x * 8 + (threadIdx.x >> 5);
  float s = 0.0f;
#pragma unroll
  for (int i = 0; i < 4; ++i) {
    int idx = lane + i * 32;
    s += head[(size_t)b * 129 + idx] * fc1_b[idx];
  }
#pragma unroll
  for (int off = 16; off > 0; off >>= 1) s += __shfl_down(s, off, 32);
  if (lane == 0) pbias[b] = s;
}

// ---------------------------------------------------------------------------
// scores[b,m] = proj[b,:] . memrow(b,m) + pbias[b] + head[b,128]*sigmoid(last_usage[b,m])
// memrow(b,0) == h_0[b,:].   One wave per (b,m); 8 waves/block; grid = B*M/8.
// mem is streamed with non-temporal loads (256MB > 192MB L2).
// ---------------------------------------------------------------------------
__global__ void scores_kernel(const float* __restrict__ proj, const float* __restrict__ pbias,
                              const float* __restrict__ head, const float* __restrict__ last_usage,
                              const float* __restrict__ mem, const float* __restrict__ h0,
                              float* __restrict__ scores)
{
  const int lane = threadIdx.x & 31;
  const int g = blockIdx.x * 8 + (threadIdx.x >> 5);
  const int b = g >> 8;                 // / MM
  const int m = g & (MM - 1);

  const float* row = (m == 0) ? (h0 + (size_t)b * HH)
                              : (mem + ((size_t)b * MM + m) * HH);
  const f4* rv = (const f4*)row;
  const f4* pv = (const f4*)(proj + (size_t)b * HH);

  f4 acc = {};
#pragma unroll
  for (int i = 0; i < 8; ++i) {
    int idx = i * 32 + lane;            // coalesced 16B per lane
    f4 mv = __builtin_nontemporal_load(rv + idx);
    f4 pr = pv[idx];
    acc += mv * pr;
  }
  float dot = acc.x + acc.y + acc.z + acc.w;
#pragma unroll
  for (int off = 16; off > 0; off >>= 1) dot += __shfl_down(dot, off, 32);

  if (lane == 0) {
    float lu = sigmoidf_(last_usage[(size_t)b * MM + m]);
    scores[(size_t)b * MM + m] = dot + pbias[b] + head[(size_t)b * 129 + 128] * lu;
  }
}

// ---------------------------------------------------------------------------
// Hard gumbel-softmax selection + entry gather.
// One block (256 threads == M) per batch row.
// x = (scores + gumbel(u)) * tau ; hard = (exp(x-xmax) == emax) ;
// entry[b,:] = sum_{m: hard} memrow(b,m).
// ---------------------------------------------------------------------------
__global__ void select_gather(const float* __restrict__ scores, const float* __restrict__ u,
                              const float* __restrict__ mem, const float* __restrict__ h0,
                              float* __restrict__ entry_out, int out_stride, float tau)
{
  __shared__ float red[MM];
  __shared__ int   cnt;
  __shared__ int   list[64];

  const int t = threadIdx.x;
  const int b = blockIdx.x;

  float x = (scores[(size_t)b * MM + t] + gumbelf_(u[(size_t)b * MM + t])) * tau;

  red[t] = x; __syncthreads();
  for (int s = MM / 2; s > 0; s >>= 1) { if (t < s) red[t] = fmaxf(red[t], red[t + s]); __syncthreads(); }
  const float xmax = red[0]; __syncthreads();

  float e = expf(x - xmax);
  red[t] = e; __syncthreads();
  for (int s = MM / 2; s > 0; s >>= 1) { if (t < s) red[t] = fmaxf(red[t], red[t + s]); __syncthreads(); }
  const float emax = red[0];

  if (t == 0) cnt = 0;
  __syncthreads();
  if (e == emax) {
    int idx = atomicAdd(&cnt, 1);
    if (idx < 64) list[idx] = t;
  }
  __syncthreads();

  const int c = cnt < 64 ? cnt : 64;
  f4 acc = {};
  for (int i = 0; i < c; ++i) {
    const int m = list[i];
    const float* row = (m == 0) ? (h0 + (size_t)b * HH)
                                : (mem + ((size_t)b * MM + m) * HH);
    acc += ((const f4*)row)[t];          // t covers 256*4 = 1024 floats
  }
  ((f4*)(entry_out + (size_t)b * out_stride))[t] = acc;
}

// ---------------------------------------------------------------------------
// Fused GRU epilogue with 3 parallel WMMA accumulators:
//   h_new = tanh(inp@Ws1 + b1 + r*(h0@Ws2 + b2) + z*(b3 + ent@Ws3))
//   h_1   = n*h_new + (1-n)*h0 ;  writes h_1 into d_out[0:B*H] and o[:, :H].
// One wave per 16x64 tile; grid = (1024/64, 256/16), blockDim = 32.
// ---------------------------------------------------------------------------
__global__ void gru_out_wmma(const float* __restrict__ inp, const float* __restrict__ Ws1,
                             const float* __restrict__ h0,  const float* __restrict__ Ws2,
                             const float* __restrict__ ent, const float* __restrict__ Ws3,
                             const float* __restrict__ b1v, const float* __restrict__ b2v,
                             const float* __restrict__ b3v, const float* __restrict__ wb,
                             float* __restrict__ out)
{
  const int lane = threadIdx.x & 31;
  const int half = lane >> 4;
  const int r    = lane & 15;
  const int m0   = blockIdx.y * 16;
  const int n0   = blockIdx.x * 64;

  v8f acc1[4] = {}, acc2[4] = {}, acc3[4] = {};

  const int arow = m0 + r;
  const float* a1p = inp + (size_t)arow * HH + 2 * half;   // 8B aligned (even offsets)
  const float* a2p = h0  + (size_t)arow * HH + 2 * half;
  const float* a3p = ent + (size_t)arow * HH + 2 * half;

  for (int k0 = 0; k0 < HH; k0 += 4) {
    v2f a1 = *(const v2f*)(a1p + k0);
    v2f a2 = *(const v2f*)(a2p + k0);
    v2f a3 = *(const v2f*)(a3p + k0);
    const int krow = k0 + 2 * half;
    const float* w1r0 = Ws1 + (size_t)krow * HH; const float* w1r1 = w1r0 + HH;
    const float* w2r0 = Ws2 + (size_t)krow * HH; const float* w2r1 = w2r0 + HH;
    const float* w3r0 = Ws3 + (size_t)krow * HH; const float* w3r1 = w3r0 + HH;
#pragma unroll
    for (int t = 0; t < 4; ++t) {
      const int c = n0 + t * 16 + r;
      v2f b1; b1.x = w1r0[c]; b1.y = w1r1[c];
      v2f b2; b2.x = w2r0[c]; b2.y = w2r1[c];
      v2f b3; b3.x = w3r0[c]; b3.y = w3r1[c];
      acc1[t] = wmma4(a1, b1, acc1[t]);
      acc2[t] = wmma4(a2, b2, acc2[t]);
      acc3[t] = wmma4(a3, b3, acc3[t]);
    }
  }

#pragma unroll
  for (int t = 0; t < 4; ++t) {
    const int col = n0 + t * 16 + r;
#pragma unroll
    for (int v = 0; v < 8; ++v) {
      const int row = m0 + v + 8 * half;
      const float s1 = acc1[t][v] + b1v[col];
      const float s2 = acc2[t][v] + b2v[col];
      const float s3 = acc3[t][v] + b3v[col];
      const float* wbrow = wb + (size_t)row * N3H;
      const float rg = wbrow[col];
      const float zg = wbrow[HH + col];
      const float ng = wbrow[2 * HH + col];
      const float hn = tanhf(s1 + rg * s2 + zg * s3);
      const float h1 = ng * hn + (1.0f - ng) * h0[(size_t)row * HH + col];
      out[(size_t)row * HH + col] = h1;                               // h_1
      out[(size_t)BB * HH + (size_t)row * (2 * HH) + col] = h1;       // o[:, :H]
    }
  }
}

// ---------------------------------------------------------------------------
extern "C" void kernel_launch(void* const* d_in, const int* in_sizes, int n_in,
                              void* d_out, int out_size, void* d_ws, size_t ws_size,
                              hipStream_t stream) {
  const float* input_     = (const float*)d_in[0];
  const float* h_0        = (const float*)d_in[1];
  const float* mem        = (const float*)d_in[2];
  const float* last_usage = (const float*)d_in[3];
  const float* u1         = (const float*)d_in[4];
  const float* u2         = (const float*)d_in[5];
  const float* W_ih       = (const float*)d_in[6];
  const float* W_hh       = (const float*)d_in[7];
  const float* W_rh       = (const float*)d_in[8];
  const float* W_s1       = (const float*)d_in[9];
  const float* W_s2       = (const float*)d_in[10];
  const float* W_s3       = (const float*)d_in[11];
  const float* bias       = (const float*)d_in[12];
  const float* W_im       = (const float*)d_in[13];
  const float* W_hm       = (const float*)d_in[14];
  const float* fc1_w      = (const float*)d_in[15];
  const float* fc1_b      = (const float*)d_in[16];
  const float* W_im1      = (const float*)d_in[17];
  const float* W_hm1      = (const float*)d_in[18];
  const float* W_mm1      = (const float*)d_in[19];
  const float* bias_m1    = (const float*)d_in[20];
  const float* bias_1     = (const float*)d_in[21];
  const float* bias_2     = (const float*)d_in[22];
  const float* bias_3     = (const float*)d_in[23];

  float* ws     = (float*)d_ws;
  float* rh     = ws;                   // B*129           (read_head / head1)
  float* proj   = rh     + BB * 129;    // B*H             (rh_proj / h1_proj)
  float* pbias  = proj   + BB * HH;     // B
  float* scores = pbias  + BB;          // B*M
  float* entry  = scores + BB * MM;     // B*H
  float* wb     = entry  + BB * HH;     // B*3H

  float* out = (float*)d_out;

  const dim3 blk32(32, 1, 1);
  const dim3 gN129((129 + 63) / 64, BB / 16, 1);
  const dim3 gN1024(HH / 64, BB / 16, 1);
  const dim3 gN3072(N3H / 64, BB / 16, 1);

  // 1. read_head = tanh(input_@W_im + h_0@W_hm)     (B x 129)
  gemm3_wmma<2, 2><<<gN129, blk32, 0, stream>>>(
      input_, W_im, h_0, W_hm, nullptr, nullptr, nullptr, rh, HH, 129, HH);
  // 2. rh_proj = read_head[:, :128] @ fc1_w          (B x H)
  gemm3_wmma<1, 0><<<gN1024, blk32, 0, stream>>>(
      rh, fc1_w, nullptr, nullptr, nullptr, nullptr, nullptr, proj, HSD, HH, 129);
  // 3. pbias[b] = read_head[b,:128] . fc1_b
  proj_bias<<<BB / 8, 256, 0, stream>>>(rh, fc1_b, pbias);
  // 4. scores pass 1 (streams mem, NT)
  scores_kernel<<<(BB * MM) / 8, 256, 0, stream>>>(
      proj, pbias, rh, last_usage, mem, h_0, scores);
  // 5. y1 selection + entry gather
  select_gather<<<BB, MM, 0, stream>>>(scores, u1, mem, h_0, entry, HH, 1.0f /*TAU*/);
  // 6. w_b = sigmoid(input_@W_ih + h_0@W_hh + entry@W_rh + bias)   (B x 3H)
  gemm3_wmma<3, 1><<<gN3072, blk32, 0, stream>>>(
      input_, W_ih, h_0, W_hh, entry, W_rh, bias, wb, HH, N3H, HH);
  // 7. fused GRU epilogue -> h_1 and o[:, :H]
  gru_out_wmma<<<gN1024, blk32, 0, stream>>>(
      input_, W_s1, h_0, W_s2, entry, W_s3, bias_1, bias_2, bias_3, wb, out);
  // 8. head1 = tanh(input_@W_im1 + h_0@W_hm1 + entry@W_mm1 + bias_m1)   (B x 129)
  gemm3_wmma<3, 2><<<gN129, blk32, 0, stream>>>(
      input_, W_im1, h_0, W_hm1, entry, W_mm1, bias_m1, rh, HH, 129, HH);
  // 9. h1_proj = head1[:, :128] @ fc1_w
  gemm3_wmma<1, 0><<<gN1024, blk32, 0, stream>>>(
      rh, fc1_w, nullptr, nullptr, nullptr, nullptr, nullptr, proj, HSD, HH, 129);
  // 10. pbias for head1
  proj_bias<<<BB / 8, 256, 0, stream>>>(rh, fc1_b, pbias);
  // 11. scores pass 2 (streams mem, NT)
  scores_kernel<<<(BB * MM) / 8, 256, 0, stream>>>(
      proj, pbias, rh, last_usage, mem, h_0, scores);
  // 12. y2 selection + entry_o gather -> o[:, H:2H]
  select_gather<<<BB, MM, 0, stream>>>(scores, u2, mem, h_0,
                                       out + (size_t)BB * HH + HH, 2 * HH, 1.0f);
}